// LDAM_meta_2336462209283
// MI455X (gfx1250) — compile-verified
//
#include <hip/hip_runtime.h>
#include <hip/hip_bf16.h>
#include <math.h>

#define C_CLS 1000
#define A_FEA 128
#define N_BAT 512

typedef __attribute__((ext_vector_type(2))) float v2f;
typedef __attribute__((ext_vector_type(8))) float v8f;

// ---------------- workspace layout (floats) ----------------
#define OFF_COUNTS   0                        // C
#define OFF_FEATSUM  (OFF_COUNTS + C_CLS)     // C*A
#define OFF_AVE      (OFF_FEATSUM + C_CLS*A_FEA) // C*A
#define OFF_SCALE    (OFF_AVE + C_CLS*A_FEA)  // C   (w/cnt_safe)
#define OFF_W        (OFF_SCALE + C_CLS)      // C
#define OFF_AUG      (OFF_W + C_CLS)          // N*C
#define OFF_WNLL     (OFF_AUG + N_BAT*C_CLS)  // N
#define OFF_WSUM     (OFF_WNLL + N_BAT)       // N

// ---------------- K0: zero the atomic accumulators ----------------
__global__ void k_zero_stats(float* ws) {
    int i = blockIdx.x * blockDim.x + threadIdx.x;
    if (i < C_CLS + C_CLS * A_FEA) ws[OFF_COUNTS + i] = 0.0f;
}

// ---------------- K1: segment sums (counts, feat_sum) ----------------
__global__ void k_seg_sums(const float* __restrict__ features,
                           const int* __restrict__ labels, float* ws) {
    int n = blockIdx.x;          // 512 blocks x 128 threads
    int tid = threadIdx.x;
    int y = labels[n];
    atomicAdd(&ws[OFF_FEATSUM + y * A_FEA + tid], features[n * A_FEA + tid]);
    if (tid == 0) atomicAdd(&ws[OFF_COUNTS + y], 1.0f);
}

// ---------------- K2: per-class stats: ave, w, new_ave, new_amount ----------------
__global__ void k_class_stats(const float* __restrict__ ave_state,
                              const float* __restrict__ amount_state,
                              float* ws, float* __restrict__ new_ave,
                              float* __restrict__ new_amount) {
    int c = blockIdx.x;          // 1000 blocks x 128 threads
    int tid = threadIdx.x;
    float cnt = ws[OFF_COUNTS + c];
    float cnt_safe = (cnt == 0.0f) ? 1.0f : cnt;
    float am = amount_state[c];
    float denom = cnt + am;
    float w = (denom > 0.0f) ? (cnt / denom) : 0.0f;
    float av = ws[OFF_FEATSUM + c * A_FEA + tid] / cnt_safe;
    ws[OFF_AVE + c * A_FEA + tid] = av;
    new_ave[c * A_FEA + tid] = ave_state[c * A_FEA + tid] * (1.0f - w) + av * w;
    if (tid == 0) {
        new_amount[c] = am + cnt;
        ws[OFF_SCALE + c] = w / cnt_safe;
        ws[OFF_W + c] = w;
    }
}

// ---------------- K3: new_cov init = cov*(1-w) + w(1-w)*diff diff^T ----------------
__global__ void k_cov_init(const float* __restrict__ cov_state,
                           const float* __restrict__ ave_state,
                           const float* ws, float* __restrict__ new_cov) {
    __shared__ float diff[A_FEA];
    int c = blockIdx.x;          // 1000 blocks x 256 threads
    int tid = threadIdx.x;
    if (tid < A_FEA)
        diff[tid] = ave_state[c * A_FEA + tid] - ws[OFF_AVE + c * A_FEA + tid];
    __syncthreads();
    float w = ws[OFF_W + c];
    float omw = 1.0f - w;
    float ww = w * omw;
    size_t base = (size_t)c * A_FEA * A_FEA;
    for (int i = tid; i < A_FEA * A_FEA; i += 256)
        new_cov[base + i] = cov_state[base + i] * omw +
                            ww * diff[i >> 7] * diff[i & (A_FEA - 1)];
}

// ---------------- K4: accumulate var*w via per-sample outer products ----------------
__global__ void k_cov_accum(const float* __restrict__ features,
                            const int* __restrict__ labels,
                            const float* ws, float* __restrict__ new_cov) {
    __shared__ float d[A_FEA];
    __shared__ float sc;
    int n = blockIdx.x;          // 512 blocks x 128 threads
    int tid = threadIdx.x;
    int y = labels[n];
    d[tid] = features[n * A_FEA + tid] - ws[OFF_AVE + y * A_FEA + tid];
    if (tid == 0) sc = ws[OFF_SCALE + y];
    __syncthreads();
    size_t base = (size_t)y * A_FEA * A_FEA;
    float s = sc;
    for (int i = tid; i < A_FEA * A_FEA; i += 128)
        atomicAdd(&new_cov[base + i], d[i >> 7] * d[i & (A_FEA - 1)] * s);
}

// ---------------- K5: meta_saug via fp32 WMMA 16x16x4 ----------------
// block = 256 threads (8 waves), 8 blocks per sample; Sigma staged once per
// block (halves L2->LDS traffic vs 16 blocks/sample). Loop order: k outer,
// a-tile inner with 4 resident accumulators, so each A fragment is loaded
// from LDS once per k-chunk instead of once per (k, a-tile).
__global__ void k_meta_saug_wmma(const float* __restrict__ W,      // C x A
                                 const float* __restrict__ y_s,    // N x C
                                 const int* __restrict__ labels,
                                 const float* __restrict__ m_list,
                                 const float* __restrict__ cv,     // C x A x A
                                 float* __restrict__ aug) {        // N x C (ws)
    extern __shared__ float lds[];
    float* sSig = lds;                       // 128*128
    float* sWy  = lds + A_FEA * A_FEA;       // 128
    float* sD   = sWy + A_FEA;               // 8 waves * 16*128

    const int n    = blockIdx.x >> 3;
    const int tg   = blockIdx.x & 7;
    const int tid  = threadIdx.x;            // 256
    const int wave = tid >> 5;               // 0..7
    const int lane = tid & 31;
    const int lo   = lane & 15;
    const int hi   = lane >> 4;
    const int y    = labels[n];

    // Stage Sigma = cv[y] (64 KB) cooperatively via 128-bit loads.
    {
        const float4* src = (const float4*)(cv + (size_t)y * A_FEA * A_FEA);
        float4* dst = (float4*)sSig;
        for (int i = tid; i < (A_FEA * A_FEA) / 4; i += 256) dst[i] = src[i];
    }
    if (tid < A_FEA) sWy[tid] = W[y * A_FEA + tid];
    __syncthreads();

    const int tile = tg * 8 + wave;          // 0..63 (63 tiles needed)
    const int c0 = tile * 16;
    float* myD = sD + wave * (16 * A_FEA);

    if (c0 < C_CLS) {
        // Build D tile (16 classes x 128) in LDS: D = W[c] - W[y]
        for (int i = lane; i < 16 * A_FEA; i += 32) {
            int r = i >> 7;
            int b = i & (A_FEA - 1);
            int c = c0 + r;
            if (c >= C_CLS) c = C_CLS - 1;   // clamp; result discarded on write
            __builtin_prefetch(&W[c * A_FEA + b], 0, 1);
            myD[i] = W[c * A_FEA + b] - sWy[b];
        }
    }
    __syncthreads();

    if (c0 < C_CLS) {
        float sig[8];
#pragma unroll
        for (int v = 0; v < 8; ++v) sig[v] = 0.0f;

        const float* dRowA = myD + lo * A_FEA;   // A frag: row m = lane&15
        const float* dFold = myD + 8 * hi * A_FEA;

#pragma unroll
        for (int half = 0; half < 2; ++half) {
            const int abase = half * 64;         // 4 a-tiles per pass
            v8f acc0 = {}, acc1 = {}, acc2 = {}, acc3 = {};
            const float* bBase = sSig + (abase + lo) * A_FEA;
#pragma unroll 4
            for (int k0 = 0; k0 < A_FEA; k0 += 4) {
                const int kk = k0 + 2 * hi;      // K = reg + 2*(lane>>4)
                v2f af;                          // A fragment: invariant over a
                af.x = dRowA[kk];
                af.y = dRowA[kk + 1];
                v2f b0, b1, b2, b3;              // B fragments: Sigma rows
                b0.x = bBase[kk];
                b0.y = bBase[kk + 1];
                b1.x = bBase[16 * A_FEA + kk];
                b1.y = bBase[16 * A_FEA + kk + 1];
                b2.x = bBase[32 * A_FEA + kk];
                b2.y = bBase[32 * A_FEA + kk + 1];
                b3.x = bBase[48 * A_FEA + kk];
                b3.y = bBase[48 * A_FEA + kk + 1];
                acc0 = __builtin_amdgcn_wmma_f32_16x16x4_f32(
                    false, af, false, b0, (short)0, acc0, false, false);
                acc1 = __builtin_amdgcn_wmma_f32_16x16x4_f32(
                    false, af, false, b1, (short)0, acc1, false, false);
                acc2 = __builtin_amdgcn_wmma_f32_16x16x4_f32(
                    false, af, false, b2, (short)0, acc2, false, false);
                acc3 = __builtin_amdgcn_wmma_f32_16x16x4_f32(
                    false, af, false, b3, (short)0, acc3, false, false);
            }
            // acc_t[v] holds T[c0 + v + 8*hi][abase + 16*t + lo]
#pragma unroll
            for (int v = 0; v < 8; ++v) {
                const float* dr = dFold + v * A_FEA + abase + lo;
                sig[v] += acc0[v] * dr[0];
                sig[v] += acc1[v] * dr[16];
                sig[v] += acc2[v] * dr[32];
                sig[v] += acc3[v] * dr[48];
            }
        }

        // Reduce across the 16 'lo' lanes of each half (classes differ by hi)
#pragma unroll
        for (int v = 0; v < 8; ++v) {
            sig[v] += __shfl_xor(sig[v], 1, 32);
            sig[v] += __shfl_xor(sig[v], 2, 32);
            sig[v] += __shfl_xor(sig[v], 4, 32);
            sig[v] += __shfl_xor(sig[v], 8, 32);
        }

        if (lo == 0) {
            float my_m = m_list[y];
#pragma unroll
            for (int v = 0; v < 8; ++v) {
                int c = c0 + v + 8 * hi;
                if (c < C_CLS) {
                    // aug = y_s + 0.5 * (RATIO * quadform) = y_s + 0.25*q
                    float val = y_s[(size_t)n * C_CLS + c] + 0.25f * sig[v];
                    if (c == y) val -= my_m;
                    aug[(size_t)n * C_CLS + c] = val;
                }
            }
        }
    }
}

// ---------------- K6: per-sample weighted NLL via log-sum-exp ----------------
__global__ void k_ce_per_sample(const float* __restrict__ aug,
                                const int* __restrict__ labels,
                                const float* __restrict__ weights,
                                float* __restrict__ wnll,
                                float* __restrict__ wsum) {
    __shared__ float red[256];
    int n = blockIdx.x;
    int tid = threadIdx.x;
    const float* row = aug + (size_t)n * C_CLS;

    float mx = -INFINITY;
    for (int c = tid; c < C_CLS; c += 256) mx = fmaxf(mx, row[c]);
    red[tid] = mx;
    __syncthreads();
    for (int s = 128; s > 0; s >>= 1) {
        if (tid < s) red[tid] = fmaxf(red[tid], red[tid + s]);
        __syncthreads();
    }
    mx = red[0];
    __syncthreads();

    float se = 0.0f;
    for (int c = tid; c < C_CLS; c += 256) se += __expf(row[c] - mx);
    red[tid] = se;
    __syncthreads();
    for (int s = 128; s > 0; s >>= 1) {
        if (tid < s) red[tid] += red[tid + s];
        __syncthreads();
    }
    if (tid == 0) {
        int y = labels[n];
        float logp = row[y] - mx - __logf(red[0]);
        float w = weights[y];
        wnll[n] = -w * logp;
        wsum[n] = w;
    }
}

// ---------------- K7: final deterministic reduction -> loss ----------------
__global__ void k_finalize_loss(const float* __restrict__ wnll,
                                const float* __restrict__ wsum,
                                float* __restrict__ out_loss) {
    __shared__ float a[512];
    __shared__ float b[512];
    int t = threadIdx.x;
    a[t] = wnll[t];
    b[t] = wsum[t];
    __syncthreads();
    for (int s = 256; s > 0; s >>= 1) {
        if (t < s) { a[t] += a[t + s]; b[t] += b[t + s]; }
        __syncthreads();
    }
    if (t == 0) out_loss[0] = a[0] / b[0];
}

extern "C" void kernel_launch(void* const* d_in, const int* in_sizes, int n_in,
                              void* d_out, int out_size, void* d_ws, size_t ws_size,
                              hipStream_t stream) {
    const float* fc_weight    = (const float*)d_in[0];
    const float* features     = (const float*)d_in[1];
    const float* y_s          = (const float*)d_in[2];
    const int*   labels       = (const int*)d_in[3];
    const float* weights      = (const float*)d_in[4];
    const float* m_list       = (const float*)d_in[5];
    const float* cv           = (const float*)d_in[6];
    const float* cov_state    = (const float*)d_in[7];
    const float* ave_state    = (const float*)d_in[8];
    const float* amount_state = (const float*)d_in[9];

    float* out        = (float*)d_out;
    float* out_loss   = out;
    float* out_cov    = out + 1;
    float* out_ave    = out_cov + (size_t)C_CLS * A_FEA * A_FEA;
    float* out_amount = out_ave + (size_t)C_CLS * A_FEA;

    float* ws = (float*)d_ws;

    // 1) zero atomic accumulators (counts + feat_sum)
    {
        int n = C_CLS + C_CLS * A_FEA;
        k_zero_stats<<<(n + 255) / 256, 256, 0, stream>>>(ws);
    }
    // 2) segment sums
    k_seg_sums<<<N_BAT, A_FEA, 0, stream>>>(features, labels, ws);
    // 3) per-class stats -> ave, w, scale, new_ave, new_amount
    k_class_stats<<<C_CLS, A_FEA, 0, stream>>>(ave_state, amount_state, ws,
                                               out_ave, out_amount);
    // 4) new_cov init
    k_cov_init<<<C_CLS, 256, 0, stream>>>(cov_state, ave_state, ws, out_cov);
    // 5) per-sample outer-product accumulation (var * w)
    k_cov_accum<<<N_BAT, A_FEA, 0, stream>>>(features, labels, ws, out_cov);
    // 6) WMMA meta_saug -> aug logits in ws
    {
        size_t shmem = (size_t)(A_FEA * A_FEA + A_FEA + 8 * 16 * A_FEA) * sizeof(float);
        k_meta_saug_wmma<<<N_BAT * 8, 256, shmem, stream>>>(
            fc_weight, y_s, labels, m_list, cv, ws + OFF_AUG);
    }
    // 7) weighted CE per sample
    k_ce_per_sample<<<N_BAT, 256, 0, stream>>>(ws + OFF_AUG, labels, weights,
                                               ws + OFF_WNLL, ws + OFF_WSUM);
    // 8) final deterministic reduction
    k_finalize_loss<<<1, N_BAT, 0, stream>>>(ws + OFF_WNLL, ws + OFF_WSUM, out_loss);
}